// GCNN_31250182045888
// MI455X (gfx1250) — compile-verified
//
#include <hip/hip_runtime.h>
#include <hip/hip_bf16.h>

typedef float v2f __attribute__((ext_vector_type(2)));
typedef float v8f __attribute__((ext_vector_type(8)));

#define N_NODES 50000
#define N_EDGES 800000
#define IN_D 64
#define HID_D 64
#define OUT_D 32

// ---------------------------------------------------------------- utilities
__global__ void gcn_zero_f32(float* p, long n) {
    long i = (long)blockIdx.x * blockDim.x + threadIdx.x;
    if (i < n) p[i] = 0.0f;
}

// degree accumulation: degA over src, degB over dst
__global__ void gcn_degrees(const int* __restrict__ src,
                            const int* __restrict__ dst,
                            float* __restrict__ degA,
                            float* __restrict__ degB, int E) {
    int e = blockIdx.x * blockDim.x + threadIdx.x;
    if (e < E) {
        atomicAdd(degA + src[e], 1.0f);
        atomicAdd(degB + dst[e], 1.0f);
    }
}

// in-place deg -> deg^{-1/2}, clamped to >= 1 (covers both arrays: n = 2*N)
__global__ void gcn_norms(float* __restrict__ deg, long n) {
    long i = (long)blockIdx.x * blockDim.x + threadIdx.x;
    if (i < n) deg[i] = rsqrtf(fmaxf(deg[i], 1.0f));
}

// feat[n][d] = h[d][n] * normSrc[n]   (h is [64][N] row-major; read coalesced)
__global__ void gcn_scale_transpose(const float* __restrict__ h,
                                    const float* __restrict__ nrmS,
                                    float* __restrict__ feat) {
    long t = (long)blockIdx.x * blockDim.x + threadIdx.x;
    if (t >= (long)IN_D * N_NODES) return;
    int d = (int)(t / N_NODES);
    int n = (int)(t % N_NODES);
    feat[(size_t)n * IN_D + d] = h[t] * nrmS[n];
}

// one wave per edge; lane carries 2 consecutive floats of the 64-wide row.
// agg rows (12.8 MB) are L2-resident -> atomics resolve in L2.
__global__ void gcn_scatter(const int* __restrict__ src,
                            const int* __restrict__ dst,
                            const float* __restrict__ feat,
                            float* __restrict__ agg, int E) {
    int wid  = (blockIdx.x * blockDim.x + threadIdx.x) >> 5;
    int lane = threadIdx.x & 31;
    if (wid >= E) return;
    int s = src[wid];
    int t = dst[wid];
    const float2 v = *(const float2*)(feat + (size_t)s * 64 + lane * 2);
    float* a = agg + (size_t)t * 64 + lane * 2;
    atomicAdd(a + 0, v.x);
    atomicAdd(a + 1, v.y);
}

// ------------------------------------------------------- WMMA GEMM, layer 1
// y = relu( (agg * nrmD) @ W1 + b1 ) * nrmS  -> feat   (pre-scaled for layer 2)
// grid.x = N/16 = 3125 row tiles, blockDim = 128 (4 waves = 4 col tiles of 16)
__global__ void gcn_gemm1_wmma(const float* __restrict__ agg,
                               const float* __restrict__ nrmD,
                               const float* __restrict__ nrmS,
                               const float* __restrict__ W1,
                               const float* __restrict__ b1,
                               float* __restrict__ feat) {
    const int lane  = threadIdx.x & 31;
    const int wv    = threadIdx.x >> 5;          // col tile 0..3
    const int m     = lane & 15;                 // A row within tile / B col
    const int khalf = (lane >> 4) << 1;          // 0 (lanes 0-15) or 2 (16-31)
    const int row   = blockIdx.x * 16 + m;       // node index for A operand
    const int col   = (wv << 4) + m;             // output column 0..63

    const float        scale = nrmD[row];
    const float* const arow  = agg + (size_t)row * 64;

    v8f acc = {};
#pragma unroll
    for (int kk = 0; kk < 64; kk += 4) {
        const int k = kk + khalf;
        v2f a, b;
        a.x = arow[k] * scale;
        a.y = arow[k + 1] * scale;
        b.x = W1[k * HID_D + col];
        b.y = W1[(k + 1) * HID_D + col];
        acc = __builtin_amdgcn_wmma_f32_16x16x4_f32(
            false, a, false, b, (short)0, acc, false, false);
    }

    // C/D layout: VGPR j -> M = j + (lane>=16 ? 8 : 0), N = lane&15
    const int   rbase = blockIdx.x * 16 + ((lane >> 4) << 3);
    const float bias  = b1[col];
#pragma unroll
    for (int j = 0; j < 8; ++j) {
        const int r = rbase + j;
        float v = acc[j] + bias;
        v = v > 0.0f ? v : 0.0f;                 // ReLU
        feat[(size_t)r * 64 + col] = v * nrmS[r]; // pre-scale for layer-2 gather
    }
}

// ------------------------------------------------------- WMMA GEMM, layer 2
// out[c][n] = ( (agg * nrmD) @ W2 + b2 )[n][c]   (transposed store, [32][N])
// grid.x = 3125, blockDim = 64 (2 waves = 2 col tiles of 16)
__global__ void gcn_gemm2_wmma(const float* __restrict__ agg,
                               const float* __restrict__ nrmD,
                               const float* __restrict__ W2,
                               const float* __restrict__ b2,
                               float* __restrict__ out) {
    const int lane  = threadIdx.x & 31;
    const int wv    = threadIdx.x >> 5;          // col tile 0..1
    const int m     = lane & 15;
    const int khalf = (lane >> 4) << 1;
    const int row   = blockIdx.x * 16 + m;
    const int col   = (wv << 4) + m;             // output column 0..31

    const float        scale = nrmD[row];
    const float* const arow  = agg + (size_t)row * 64;

    v8f acc = {};
#pragma unroll
    for (int kk = 0; kk < 64; kk += 4) {
        const int k = kk + khalf;
        v2f a, b;
        a.x = arow[k] * scale;
        a.y = arow[k + 1] * scale;
        b.x = W2[k * OUT_D + col];
        b.y = W2[(k + 1) * OUT_D + col];
        acc = __builtin_amdgcn_wmma_f32_16x16x4_f32(
            false, a, false, b, (short)0, acc, false, false);
    }

    const int   rbase = blockIdx.x * 16 + ((lane >> 4) << 3);
    const float bias  = b2[col];
#pragma unroll
    for (int j = 0; j < 8; ++j) {
        const int r = rbase + j;
        out[(size_t)col * N_NODES + r] = acc[j] + bias;
    }
}

// ----------------------------------------------------------------- launcher
extern "C" void kernel_launch(void* const* d_in, const int* in_sizes, int n_in,
                              void* d_out, int out_size, void* d_ws, size_t ws_size,
                              hipStream_t stream) {
    const float* h   = (const float*)d_in[0];  // [64][N]
    const int*   src = (const int*)  d_in[1];  // [E]
    const int*   dst = (const int*)  d_in[2];  // [E]
    const float* W1  = (const float*)d_in[3];  // [64][64]
    const float* b1  = (const float*)d_in[4];  // [64]
    const float* W2  = (const float*)d_in[5];  // [64][32]
    const float* b2  = (const float*)d_in[6];  // [32]
    float*       out = (float*)d_out;          // [32][N]

    // workspace layout (floats): nrmS[N] | nrmD[N] | feat[64N] | agg[64N]
    float* nrmS = (float*)d_ws;
    float* nrmD = nrmS + N_NODES;
    float* feat = nrmD + N_NODES;
    float* agg  = feat + (size_t)N_NODES * 64;

    const int  T    = 256;
    const long degN = 2L * N_NODES;
    const long aggN = 64L * N_NODES;

    // degrees -> norms
    gcn_zero_f32<<<(unsigned)((degN + T - 1) / T), T, 0, stream>>>(nrmS, degN);
    gcn_degrees <<<(N_EDGES + T - 1) / T, T, 0, stream>>>(src, dst, nrmS, nrmD, N_EDGES);
    gcn_norms   <<<(unsigned)((degN + T - 1) / T), T, 0, stream>>>(nrmS, degN);

    // layer 1: scaled transpose, scatter-add, WMMA GEMM (+ReLU, +src prescale)
    gcn_scale_transpose<<<(unsigned)((aggN + T - 1) / T), T, 0, stream>>>(h, nrmS, feat);
    gcn_zero_f32<<<(unsigned)((aggN + T - 1) / T), T, 0, stream>>>(agg, aggN);
    gcn_scatter <<<(N_EDGES * 32 + T - 1) / T, T, 0, stream>>>(src, dst, feat, agg, N_EDGES);
    gcn_gemm1_wmma<<<N_NODES / 16, 128, 0, stream>>>(agg, nrmD, nrmS, W1, b1, feat);

    // layer 2: scatter-add, WMMA GEMM, transposed store
    gcn_zero_f32<<<(unsigned)((aggN + T - 1) / T), T, 0, stream>>>(agg, aggN);
    gcn_scatter <<<(N_EDGES * 32 + T - 1) / T, T, 0, stream>>>(src, dst, feat, agg, N_EDGES);
    gcn_gemm2_wmma<<<N_NODES / 16, 64, 0, stream>>>(agg, nrmD, W2, b2, out);
}